// DotProductAttention_12317966205049
// MI455X (gfx1250) — compile-verified
//
#include <hip/hip_runtime.h>

typedef __attribute__((ext_vector_type(2))) float v2f;
typedef __attribute__((ext_vector_type(4))) float v4f;
typedef __attribute__((ext_vector_type(8))) float v8f;

#define B_BATCH 16
#define Q_LEN   2048
#define K_LEN   2048
#define D_HEAD  128
#define WAVES   8
#define TILE_Q  16
#define TILE_K  16
#define LDS_STRIDE 132   // 128 + 4 pad: lanes 0..15 hit distinct banks (4k mod 64)

__launch_bounds__(WAVES * 32, 1)
__global__ void attn_fa_wmma_f32(const float* __restrict__ Qg,
                                 const float* __restrict__ Kg,
                                 const float* __restrict__ Vg,
                                 const int*   __restrict__ valid_lens,
                                 float* __restrict__ Og)
{
    __shared__ float ldsK[TILE_K * LDS_STRIDE];
    __shared__ float ldsV[TILE_K * LDS_STRIDE];

    const int tid  = threadIdx.x;
    const int wave = tid >> 5;
    const int lane = tid & 31;
    const int lq   = lane & 15;   // q column (PV) / key row (S^T) selector
    const int hi   = lane >> 4;   // half-wave id: selects K-slot pairs {0,1} vs {2,3}
    const bool lowh = (hi == 0);

    const int b  = blockIdx.y;
    const int q0 = blockIdx.x * (WAVES * TILE_Q) + wave * TILE_Q;

    const int valid   = valid_lens[b];
    const int nchunks = (valid + TILE_K - 1) / TILE_K;   // uniform per block

    // ---- Q tile into registers, laid out as the B operand (Q^T, 4xD chunks) ----
    // lane L(<16): Q[q0+L][4c + 0..1]; lane 16+L: Q[q0+L][4c + 2..3]
    const float* qrow = Qg + ((size_t)b * Q_LEN + q0 + lq) * D_HEAD + hi * 2;
    v2f qreg[32];
#pragma unroll
    for (int c = 0; c < 32; ++c)
        qreg[c] = *(const v2f*)(qrow + c * 4);

    // ---- O^T accumulators: 8 d-tiles of 16x16 f32 (8 VGPRs each) ----
    v8f acc[8];
    const v8f vzero = {0.f, 0.f, 0.f, 0.f, 0.f, 0.f, 0.f, 0.f};
#pragma unroll
    for (int t = 0; t < 8; ++t) acc[t] = vzero;

    float m = -INFINITY;   // running row max (row = q = lane&15)
    float l = 0.0f;        // running row sum

    // ---- cooperative K/V tile prefetch: 256 thr x 8 floats = 16x128 tile ----
    const int prow = tid >> 4;         // 0..15 (key row in tile)
    const int pcol = (tid & 15) * 8;   // 0..120
    const float* kbase = Kg + (size_t)b * K_LEN * D_HEAD;
    const float* vbase = Vg + (size_t)b * K_LEN * D_HEAD;

    v4f pk0, pk1, pv0, pv1;
    {
        const float* kp = kbase + (size_t)prow * D_HEAD + pcol;
        const float* vp = vbase + (size_t)prow * D_HEAD + pcol;
        pk0 = *(const v4f*)kp;  pk1 = *(const v4f*)(kp + 4);
        pv0 = *(const v4f*)vp;  pv1 = *(const v4f*)(vp + 4);
    }

    const float inv_sqrt_d = 0.08838834764831843f;  // 1/sqrt(128)

    for (int ci = 0; ci < nchunks; ++ci) {
        const int k0 = ci * TILE_K;

        __syncthreads();  // previous chunk's LDS reads complete
        {
            float* kd = &ldsK[prow * LDS_STRIDE + pcol];
            float* vd = &ldsV[prow * LDS_STRIDE + pcol];
            *(v4f*)kd = pk0;  *(v4f*)(kd + 4) = pk1;
            *(v4f*)vd = pv0;  *(v4f*)(vd + 4) = pv1;
        }
        __syncthreads();  // tile visible to all waves

        // prefetch next chunk (global loads overlap compute below)
        if (ci + 1 < nchunks) {
            const int kn = (ci + 1) * TILE_K;
            const float* kp = kbase + (size_t)(kn + prow) * D_HEAD + pcol;
            const float* vp = vbase + (size_t)(kn + prow) * D_HEAD + pcol;
            pk0 = *(const v4f*)kp;  pk1 = *(const v4f*)(kp + 4);
            pv0 = *(const v4f*)vp;  pv1 = *(const v4f*)(vp + 4);
        }

        // ---- S^T (16 keys x 16 q) = K_tile(16x4) x Q^T(4x16), 32 K-chunks ----
        // two independent accumulation chains to avoid a 32-deep WMMA RAW chain
        v8f s0 = vzero, s1 = vzero;
#pragma unroll
        for (int c = 0; c < 32; c += 2) {
            v2f ka0 = *(const v2f*)&ldsK[lq * LDS_STRIDE + (c    ) * 4 + hi * 2];
            v2f ka1 = *(const v2f*)&ldsK[lq * LDS_STRIDE + (c + 1) * 4 + hi * 2];
            s0 = __builtin_amdgcn_wmma_f32_16x16x4_f32(
                     false, ka0, false, qreg[c],     (short)0, s0, false, false);
            s1 = __builtin_amdgcn_wmma_f32_16x16x4_f32(
                     false, ka1, false, qreg[c + 1], (short)0, s1, false, false);
        }
        v8f s = s0 + s1;

        // ---- scale, floor, mask; per-lane key rows r(+8*hi), q = lane&15 ----
        float p[8];
        float cmax = -INFINITY;
#pragma unroll
        for (int r = 0; r < 8; ++r) {
            const int kidx = k0 + r + 8 * hi;
            float sv = floorf(s[r] * inv_sqrt_d);
            sv = (kidx < valid) ? sv : -1.0e6f;
            p[r] = sv;
            cmax = fmaxf(cmax, sv);
        }
        cmax = fmaxf(cmax, __shfl_xor(cmax, 16));   // combine the two key-halves

        const float mnew  = fmaxf(m, cmax);
        const float alpha = __expf(m - mnew);
        m = mnew;

        float lsum = 0.0f;
#pragma unroll
        for (int r = 0; r < 8; ++r) {
            p[r] = __expf(p[r] - mnew);
            lsum += p[r];
        }
        lsum += __shfl_xor(lsum, 16);
        l = l * alpha + lsum;

        // rescale O^T accumulators; vector form -> v_pk_mul_f32 pairs
#pragma unroll
        for (int t = 0; t < 8; ++t)
            acc[t] = acc[t] * alpha;

        // ---- build P^T B-operands: half-swap keys 8..15 <-> 0..7 ----
        float psw[8];
#pragma unroll
        for (int r = 0; r < 8; ++r) psw[r] = __shfl_xor(p[r], 16);

        v2f pb0, pb1, pb2, pb3;
        pb0.x = lowh ? p[0]   : psw[2];  pb0.y = lowh ? p[1]   : psw[3]; // keys 0-3
        pb1.x = lowh ? p[4]   : psw[6];  pb1.y = lowh ? p[5]   : psw[7]; // keys 4-7
        pb2.x = lowh ? psw[0] : p[2];    pb2.y = lowh ? psw[1] : p[3];   // keys 8-11
        pb3.x = lowh ? psw[4] : p[6];    pb3.y = lowh ? psw[5] : p[7];   // keys 12-15

        // ---- O^T += V^T(16d x 4k) x P^T(4k x 16q), per d-tile, 4 key-subs ----
        // 8 independent accumulation chains (one per d-tile)
#pragma unroll
        for (int t = 0; t < 8; ++t) {
#pragma unroll
            for (int ksub = 0; ksub < 4; ++ksub) {
                const int krow = 4 * ksub + hi * 2;
                v2f va;
                va.x = ldsV[(krow    ) * LDS_STRIDE + t * 16 + lq];
                va.y = ldsV[(krow + 1) * LDS_STRIDE + t * 16 + lq];
                const v2f pb = (ksub == 0) ? pb0 : (ksub == 1) ? pb1
                             : (ksub == 2) ? pb2 : pb3;
                acc[t] = __builtin_amdgcn_wmma_f32_16x16x4_f32(
                             false, va, false, pb, (short)0, acc[t], false, false);
            }
        }
    }

    // ---- normalize and store: lane writes 8 consecutive d per tile ----
    const float invl = 1.0f / l;
    float* orow = Og + ((size_t)b * Q_LEN + q0 + lq) * D_HEAD + hi * 8;
#pragma unroll
    for (int t = 0; t < 8; ++t) {
        v8f o = acc[t] * invl;     // vector scale -> v_pk_mul_f32
        v4f o0, o1;
        o0.x = o[0];  o0.y = o[1];  o0.z = o[2];  o0.w = o[3];
        o1.x = o[4];  o1.y = o[5];  o1.z = o[6];  o1.w = o[7];
        *(v4f*)(orow + t * 16)     = o0;
        *(v4f*)(orow + t * 16 + 4) = o1;
    }
}

extern "C" void kernel_launch(void* const* d_in, const int* in_sizes, int n_in,
                              void* d_out, int out_size, void* d_ws, size_t ws_size,
                              hipStream_t stream)
{
    (void)in_sizes; (void)n_in; (void)out_size; (void)d_ws; (void)ws_size;
    const float* Qg = (const float*)d_in[0];
    const float* Kg = (const float*)d_in[1];
    const float* Vg = (const float*)d_in[2];
    const int*   vl = (const int*)d_in[3];
    float*       Og = (float*)d_out;

    dim3 grid(Q_LEN / (WAVES * TILE_Q), B_BATCH);   // (16, 16)
    dim3 block(WAVES * 32);                          // 256 threads = 8 waves
    attn_fa_wmma_f32<<<grid, block, 0, stream>>>(Qg, Kg, Vg, vl, Og);
}